// SwapModule_18957985644708
// MI455X (gfx1250) — compile-verified
//
#include <hip/hip_runtime.h>
#include <hip/hip_bf16.h>

typedef float v2f __attribute__((ext_vector_type(2)));
typedef float v8f __attribute__((ext_vector_type(8)));

// ---------------------------------------------------------------------------
// Stage A: per-(b,c) spatial mean over 256x256, float4 loads, LDS reduction.
// grid = 512 (one block per plane), block = 256.
// ---------------------------------------------------------------------------
__global__ void stageA_mean(const float* __restrict__ x, float* __restrict__ mean) {
  __shared__ float red[256];
  const int plane = blockIdx.x;            // b*64 + c
  const int t = threadIdx.x;
  const float4* __restrict__ x4 = (const float4*)x;
  const long base = (long)plane * 16384;   // 65536 floats = 16384 float4
  float s = 0.f;
#pragma unroll 4
  for (int q = 0; q < 64; ++q) {
    float4 v = x4[base + (q << 8) + t];
    s += v.x + v.y + v.z + v.w;
  }
  red[t] = s;
  __syncthreads();
  for (int off = 128; off > 0; off >>= 1) {
    if (t < off) red[t] += red[t + off];
    __syncthreads();
  }
  if (t == 0) mean[plane] = red[0] * (1.0f / 65536.0f);
}

// ---------------------------------------------------------------------------
// Stage B: y = mean @ fc_w^T + fc_b (V_WMMA_F32_16X16X4_F32), leaky-ReLU,
// exact top-k(32) rank map. Single wave32 (EXEC all ones, as WMMA requires).
//   D[m][n] = sum_k A[m][k] * B[k][n],  A[m][k]=mean[m][k],  B[k][n]=fc_w[n][k]
// A 16x4 layout: lanes 0-15 M=0..15 {VGPR0=K0, VGPR1=K1}; lanes 16-31 {K2,K3}
// B 4x16 layout (mirrored): VGPR0 = rows {0,2}, VGPR1 = rows {1,3}
// D 16x16: VGPR r -> M=r (lanes 0-15) / M=r+8 (lanes 16-31), N = lane%16
// ---------------------------------------------------------------------------
__global__ void stageB_fc_topk(const float* __restrict__ mean,
                               const float* __restrict__ fcw,
                               const float* __restrict__ fcb,
                               int* __restrict__ rankmap) {
  __shared__ float ysh[512];               // y[b*64 + c], b<8
  const int lane = threadIdx.x;            // 0..31
  const int half = lane >> 4;              // 0 / 1
  const int lm   = lane & 15;

  v8f acc[4];
#pragma unroll
  for (int n = 0; n < 4; ++n) acc[n] = (v8f){0.f,0.f,0.f,0.f,0.f,0.f,0.f,0.f};

  for (int kk = 0; kk < 16; ++kk) {
    const int k0 = kk * 4;
    v2f a;
    // rows 8..15 of A are zero padding (B=8)
    a.x = (lm < 8) ? mean[lm * 64 + k0 + 2 * half + 0] : 0.f;
    a.y = (lm < 8) ? mean[lm * 64 + k0 + 2 * half + 1] : 0.f;
#pragma unroll
    for (int n = 0; n < 4; ++n) {
      v2f bf;
      const float* wrow = fcw + (n * 16 + lm) * 64 + k0 + 2 * half;
      bf.x = wrow[0];
      bf.y = wrow[1];
      acc[n] = __builtin_amdgcn_wmma_f32_16x16x4_f32(
          false, a, false, bf, (short)0, acc[n], false, false);
    }
  }

  // Write y (rows 0..7 live in lanes 0..15, component r = batch r)
  if (lane < 16) {
#pragma unroll
    for (int n = 0; n < 4; ++n) {
      const int c = n * 16 + lane;
      const float bias = fcb[c];
#pragma unroll
      for (int r = 0; r < 8; ++r) {
        float v = acc[n][r] + bias;
        v = (v >= 0.f) ? v : 0.01f * v;    // LeakyReLU(0.01)
        ysh[r * 64 + c] = v;
      }
    }
  }
  __syncthreads();

  // Exact top-k rank (matches jax.lax.top_k descending, lower index on ties).
#pragma unroll
  for (int h = 0; h < 2; ++h) {
    const int c = lane + h * 32;
    for (int b = 0; b < 8; ++b) {
      const float yv = ysh[b * 64 + c];
      int rank = 0;
      for (int c2 = 0; c2 < 64; ++c2) {
        const float y2 = ysh[b * 64 + c2];
        rank += (y2 > yv) || (y2 == yv && c2 < c);
      }
      rankmap[b * 64 + c] = (rank < 32) ? rank : -1;
    }
  }
}

// ---------------------------------------------------------------------------
// Stage C: fused copy + Gaussian warp. One block = 1024 contiguous pixels of
// one (b,c) plane; x read exactly once; all stores are B128.
// grid = 8*64*64 = 32768, block = 256.
// d_out layout (float4 units): out 12582912 | exPx,exPy,sigmax,sigmay 4194304 ea
// ---------------------------------------------------------------------------
__global__ void stageC_main(const float* __restrict__ x,
                            const int* __restrict__ rankmap,
                            const float* __restrict__ wox, const float* __restrict__ box,
                            const float* __restrict__ woy, const float* __restrict__ boy,
                            const float* __restrict__ wsx, const float* __restrict__ bsx,
                            const float* __restrict__ wsy, const float* __restrict__ bsy,
                            float* __restrict__ out) {
  const int t   = threadIdx.x;
  const int seg = blockIdx.x & 63;
  const int c   = (blockIdx.x >> 6) & 63;
  const int b   = blockIdx.x >> 12;
  const int p4  = (seg << 8) + t;                   // float4 index in plane

  const float4* __restrict__ x4 = (const float4*)x;
  float4* __restrict__ o4 = (float4*)out;

  const float4 v = x4[((long)b * 64 + c) * 16384 + p4];
  o4[((long)b * 96 + c) * 16384 + p4] = v;          // out[:, :64] = x

  const int r = rankmap[b * 64 + c];                // uniform per block
  if (r < 0) return;

  const float cwox = wox[r], cbox = box[r];
  const float cwoy = woy[r], cboy = boy[r];
  const float cwsx = wsx[r], cbsx = bsx[r];
  const float cwsy = wsy[r], cbsy = bsy[r];

  const int p   = p4 << 2;                          // first pixel linear index
  const float fi  = (float)(p >> 8);                // dim-2 coordinate (row)
  const float fj0 = (float)(p & 255);               // dim-3 coordinate (col)

  const float vv[4] = {v.x, v.y, v.z, v.w};
  float sw[4], px[4], py[4], sx[4], sy[4];
#pragma unroll
  for (int e = 0; e < 4; ++e) {
    const float t0 = vv[e];
    const float ox  = 255.0f / (1.0f + __expf(-(t0 * cwox + cbox)));  // sigmoid*(W-1)
    const float oy  = 255.0f / (1.0f + __expf(-(t0 * cwoy + cboy)));
    const float sxv = fabsf(t0 * cwsx + cbsx);
    const float syv = fabsf(t0 * cwsy + cbsy);
    const float di = fi - ox;
    const float dj = (fj0 + (float)e) - oy;
    const float g  = __expf(-(di * di / (2.0f * sxv * sxv + 1e-6f) +
                              dj * dj / (2.0f * syv * syv + 1e-6f)));
    sw[e] = t0 * g; px[e] = ox; py[e] = oy; sx[e] = sxv; sy[e] = syv;
  }

  o4[((long)b * 96 + 64 + r) * 16384 + p4] = make_float4(sw[0], sw[1], sw[2], sw[3]);

  const long auxPlane = ((long)b * 32 + r) * 16384 + p4;
  const long base0 = 12582912;                      // 8*96*16384
  const long auxSz = 4194304;                       // 8*32*16384
  o4[base0 + 0 * auxSz + auxPlane] = make_float4(px[0], px[1], px[2], px[3]);
  o4[base0 + 1 * auxSz + auxPlane] = make_float4(py[0], py[1], py[2], py[3]);
  o4[base0 + 2 * auxSz + auxPlane] = make_float4(sx[0], sx[1], sx[2], sx[3]);
  o4[base0 + 3 * auxSz + auxPlane] = make_float4(sy[0], sy[1], sy[2], sy[3]);
}

// ---------------------------------------------------------------------------
extern "C" void kernel_launch(void* const* d_in, const int* in_sizes, int n_in,
                              void* d_out, int out_size, void* d_ws, size_t ws_size,
                              hipStream_t stream) {
  const float* x   = (const float*)d_in[0];
  const float* fcw = (const float*)d_in[1];
  const float* fcb = (const float*)d_in[2];
  const float* wox = (const float*)d_in[3];
  const float* box = (const float*)d_in[4];
  const float* woy = (const float*)d_in[5];
  const float* boy = (const float*)d_in[6];
  const float* wsx = (const float*)d_in[7];
  const float* bsx = (const float*)d_in[8];
  const float* wsy = (const float*)d_in[9];
  const float* bsy = (const float*)d_in[10];

  float* mean   = (float*)d_ws;                          // 512 floats
  int*   rankmp = (int*)((char*)d_ws + 512 * sizeof(float)); // 512 ints

  stageA_mean<<<512, 256, 0, stream>>>(x, mean);
  stageB_fc_topk<<<1, 32, 0, stream>>>(mean, fcw, fcb, rankmp);
  stageC_main<<<32768, 256, 0, stream>>>(x, rankmp,
                                         wox, box, woy, boy,
                                         wsx, bsx, wsy, bsy,
                                         (float*)d_out);
}